// LDGNNetwork_35905926594823
// MI455X (gfx1250) — compile-verified
//
#include <hip/hip_runtime.h>
#include <hip/hip_bf16.h>

// ---------------- problem constants ----------------
#define NN      20000      // nodes
#define EE      320000     // edges (before self loops)
#define ETOT    (EE + NN)  // edges incl. self loops
#define BB      128
#define IN_DIM  64
#define HID     64
#define HEADS   4
#define DD      256        // HID*HEADS
#define CAT     576        // HID + 2*DD
#define ENC_H   512
#define DUEL_H  128

typedef __attribute__((ext_vector_type(2))) float v2f;
typedef __attribute__((ext_vector_type(8))) float v8f;

// ---------------- small utility kernels ----------------
__global__ void zero_i32(int* p, int n) {
    for (int i = blockIdx.x * blockDim.x + threadIdx.x; i < n; i += gridDim.x * blockDim.x)
        p[i] = 0;
}

__global__ void copy_i32(const int* __restrict__ src, int* __restrict__ dst, int n) {
    for (int i = blockIdx.x * blockDim.x + threadIdx.x; i < n; i += gridDim.x * blockDim.x)
        dst[i] = src[i];
}

// count incoming edges per destination node (self loops appended logically)
__global__ void count_edges(const int* __restrict__ edst, int* __restrict__ cnt) {
    int e = blockIdx.x * blockDim.x + threadIdx.x;
    if (e >= ETOT) return;
    int d = (e < EE) ? edst[e] : (e - EE);
    atomicAdd(&cnt[d], 1);
}

// single-block exclusive scan over cnt -> row_ptr (N+1 entries)
__global__ void scan_counts(const int* __restrict__ cnt, int* __restrict__ row_ptr) {
    __shared__ int lds[1024];
    const int T = 1024;
    int t = threadIdx.x;
    const int chunk = (NN + T - 1) / T;
    int lo = t * chunk, hi = min(lo + chunk, NN);
    int s = 0;
    for (int i = lo; i < hi; ++i) s += cnt[i];
    lds[t] = s;
    __syncthreads();
    for (int off = 1; off < T; off <<= 1) {
        int v = (t >= off) ? lds[t - off] : 0;
        __syncthreads();
        lds[t] += v;
        __syncthreads();
    }
    int run = (t == 0) ? 0 : lds[t - 1];
    for (int i = lo; i < hi; ++i) { row_ptr[i] = run; run += cnt[i]; }
    if (t == T - 1) row_ptr[NN] = lds[T - 1];
}

__global__ void fill_csr(const int* __restrict__ edst, int* __restrict__ cursor,
                         int* __restrict__ col) {
    int e = blockIdx.x * blockDim.x + threadIdx.x;
    if (e >= ETOT) return;
    int d = (e < EE) ? edst[e] : (e - EE);
    int pos = atomicAdd(&cursor[d], 1);
    col[pos] = e;
}

// ---------------- WMMA fp32 GEMM:  C = act(A[M,K] @ W[K,N] + bias[N]) ----------------
// One wave computes a 16(M) x 64(N) strip of C: 4 independent accumulators, the
// A-fragment is loaded once per k-step and reused by 4 back-to-back WMMAs
// (no D->A/B hazards between them; ~6.4 FLOP/B from L2 instead of ~4).
// Fragment layouts (ISA 7.12.2, f32 16x16x4):
//   A (16x4): lane l, reg r -> M = l%16, K = 2*(l/16)+r           (documented)
//   B (4x16): lane l, reg r -> N = l%16, K = 2*(l/16)+r           (mirror of A, assumed)
//   C/D     : reg r, lane l -> M = r + 8*(l/16), N = l%16         (documented)
__global__ void gemm_wmma_f32(const float* __restrict__ A, const float* __restrict__ W,
                              const float* __restrict__ bias, float* __restrict__ C,
                              int M, int Nc, int K, int do_relu) {
    int wave = threadIdx.x >> 5;
    int lane = threadIdx.x & 31;
    int ntn = Nc >> 6;                       // tiles of 64 columns
    int total = (M >> 4) * ntn;
    int tile = blockIdx.x * 4 + wave;
    if (tile >= total) return;               // wave-uniform: EXEC stays all-ones
    int tm = tile / ntn, tn = tile - tm * ntn;
    int m0 = tm << 4, n0 = tn << 6;
    int lrow = lane & 15, lh = lane >> 4;

    v8f acc0 = {}, acc1 = {}, acc2 = {}, acc3 = {};
    const float* arow = A + (size_t)(m0 + lrow) * K + 2 * lh;  // +k0 -> K = k0+2*lh(+1)
    const float* brow = W + (size_t)(2 * lh) * Nc + n0 + lrow; // +k0*Nc -> K = k0+2*lh

    for (int k0 = 0; k0 < K; k0 += 4) {
        v2f a = { arow[k0], arow[k0 + 1] };
        const float* b0 = brow + (size_t)k0 * Nc;   // K = k0 + 2*lh
        const float* b1 = b0 + Nc;                  // K = k0 + 2*lh + 1
        v2f bb0 = { b0[0],  b1[0]  };
        v2f bb1 = { b0[16], b1[16] };
        v2f bb2 = { b0[32], b1[32] };
        v2f bb3 = { b0[48], b1[48] };
        acc0 = __builtin_amdgcn_wmma_f32_16x16x4_f32(false, a, false, bb0, (short)0, acc0, false, false);
        acc1 = __builtin_amdgcn_wmma_f32_16x16x4_f32(false, a, false, bb1, (short)0, acc1, false, false);
        acc2 = __builtin_amdgcn_wmma_f32_16x16x4_f32(false, a, false, bb2, (short)0, acc2, false, false);
        acc3 = __builtin_amdgcn_wmma_f32_16x16x4_f32(false, a, false, bb3, (short)0, acc3, false, false);
    }

    float bv0 = bias[n0 + lrow];
    float bv1 = bias[n0 + 16 + lrow];
    float bv2 = bias[n0 + 32 + lrow];
    float bv3 = bias[n0 + 48 + lrow];
#pragma unroll
    for (int r = 0; r < 8; ++r) {
        float* crow = C + (size_t)(m0 + r + 8 * lh) * Nc + n0 + lrow;
        float v0 = acc0[r] + bv0;
        float v1 = acc1[r] + bv1;
        float v2 = acc2[r] + bv2;
        float v3 = acc3[r] + bv3;
        if (do_relu) {
            v0 = v0 > 0.f ? v0 : 0.f;
            v1 = v1 > 0.f ? v1 : 0.f;
            v2 = v2 > 0.f ? v2 : 0.f;
            v3 = v3 > 0.f ? v3 : 0.f;
        }
        crow[0]  = v0;
        crow[16] = v1;
        crow[32] = v2;
        crow[48] = v3;
    }
}

// ---------------- GATv2 edge scores ----------------
// score[e][h] = sum_d att[h*64+d] * leakyrelu(xl[src][h*64+d] + xr[dst][h*64+d], 0.2)
// one wave per edge; lane handles 8 contiguous channels -> head = lane/8;
// reduce within 8-lane groups via shfl_xor.
__global__ void edge_scores(const float* __restrict__ xl, const float* __restrict__ xr,
                            const float* __restrict__ att,
                            const int* __restrict__ esrc, const int* __restrict__ edst,
                            float* __restrict__ score) {
    int wave = threadIdx.x >> 5, lane = threadIdx.x & 31;
    int e = blockIdx.x * 8 + wave;
    if (e >= ETOT) return;
    int s, d;
    if (e < EE) { s = esrc[e]; d = edst[e]; } else { s = e - EE; d = s; }
    const float* pl = xl + (size_t)s * DD;
    const float* pr = xr + (size_t)d * DD;
    int c0 = lane * 8;
    float partial = 0.f;
#pragma unroll
    for (int k = 0; k < 8; ++k) {
        float v = pl[c0 + k] + pr[c0 + k];
        v = v > 0.f ? v : 0.2f * v;
        partial += att[c0 + k] * v;
    }
    partial += __shfl_xor(partial, 1);
    partial += __shfl_xor(partial, 2);
    partial += __shfl_xor(partial, 4);
    if ((lane & 7) == 0) score[e * HEADS + (lane >> 3)] = partial;
}

// ---------------- per-node softmax + weighted aggregation (gather, no atomics) -------
// one wave per node; lane owns 8 contiguous channels (head = lane/8).
__global__ void aggregate(const int* __restrict__ row_ptr, const int* __restrict__ col,
                          const int* __restrict__ esrc, const float* __restrict__ score,
                          const float* __restrict__ xl, const float* __restrict__ bias,
                          float* __restrict__ out) {
    int wave = threadIdx.x >> 5, lane = threadIdx.x & 31;
    int node = blockIdx.x * 8 + wave;
    if (node >= NN) return;
    int head = lane >> 3;
    int beg = row_ptr[node], end = row_ptr[node + 1];

    float m = -3.402823e38f;
    for (int j = beg; j < end; ++j)
        m = fmaxf(m, score[col[j] * HEADS + head]);

    float denom = 0.f;
    for (int j = beg; j < end; ++j)
        denom += __expf(score[col[j] * HEADS + head] - m);
    float inv = 1.f / (denom + 1e-16f);

    float acc[8] = {0.f, 0.f, 0.f, 0.f, 0.f, 0.f, 0.f, 0.f};
    int c0 = lane * 8;
    for (int j = beg; j < end; ++j) {
        int e = col[j];
        float a = __expf(score[e * HEADS + head] - m) * inv;
        int s = (e < EE) ? esrc[e] : (e - EE);
        const float4* px = (const float4*)(xl + (size_t)s * DD + c0);
        float4 v0 = px[0], v1 = px[1];
        acc[0] += a * v0.x; acc[1] += a * v0.y; acc[2] += a * v0.z; acc[3] += a * v0.w;
        acc[4] += a * v1.x; acc[5] += a * v1.y; acc[6] += a * v1.z; acc[7] += a * v1.w;
    }
#pragma unroll
    for (int k = 0; k < 8; ++k) {
        float v = acc[k] + bias[c0 + k];
        out[(size_t)node * DD + c0 + k] = v > 0.f ? v : 0.f;   // reference applies relu after conv
    }
}

// ---------------- gather feat = [h_enc | g1 | g2][indices]  -> [B, 576] ----------------
__global__ void build_feat(const int* __restrict__ indices, const float* __restrict__ henc,
                           const float* __restrict__ g1, const float* __restrict__ g2,
                           float* __restrict__ feat) {
    int idx = blockIdx.x * blockDim.x + threadIdx.x;
    if (idx >= BB * CAT) return;
    int b = idx / CAT, c = idx - b * CAT;
    int n = indices[b];
    float v;
    if (c < HID)            v = henc[(size_t)n * HID + c];
    else if (c < HID + DD)  v = g1[(size_t)n * DD + (c - HID)];
    else                    v = g2[(size_t)n * DD + (c - HID - DD)];
    feat[idx] = v;
}

// ---------------- final dueling head: out = q - mean(q) + v ----------------
__global__ void head_final(const float* __restrict__ qh, const float* __restrict__ vh,
                           const float* __restrict__ qw2, const float* __restrict__ qb2,
                           const float* __restrict__ vw2, const float* __restrict__ vb2,
                           float* __restrict__ out) {
    int b = threadIdx.x;
    if (b >= BB) return;
    float q0 = qb2[0], q1 = qb2[1], v0 = vb2[0];
    const float* qr = qh + b * DUEL_H;
    const float* vr = vh + b * DUEL_H;
#pragma unroll 4
    for (int k = 0; k < DUEL_H; ++k) {
        float qv = qr[k];
        q0 += qv * qw2[k * 2 + 0];
        q1 += qv * qw2[k * 2 + 1];
        v0 += vr[k] * vw2[k];
    }
    float mean = 0.5f * (q0 + q1);
    out[b * 2 + 0] = q0 - mean + v0;
    out[b * 2 + 1] = q1 - mean + v0;
}

// ---------------- launcher ----------------
extern "C" void kernel_launch(void* const* d_in, const int* in_sizes, int n_in,
                              void* d_out, int out_size, void* d_ws, size_t ws_size,
                              hipStream_t stream) {
    const float* x       = (const float*)d_in[0];
    const int*   eidx    = (const int*)d_in[1];   // [2,E]: row0 = src, row1 = dst
    const int*   indices = (const int*)d_in[2];
    const float* enc_w1  = (const float*)d_in[3];
    const float* enc_b1  = (const float*)d_in[4];
    const float* enc_w2  = (const float*)d_in[5];
    const float* enc_b2  = (const float*)d_in[6];
    const float* wl1     = (const float*)d_in[7];
    const float* bl1     = (const float*)d_in[8];
    const float* wr1     = (const float*)d_in[9];
    const float* br1     = (const float*)d_in[10];
    const float* att1    = (const float*)d_in[11];
    const float* bias1   = (const float*)d_in[12];
    const float* wl2     = (const float*)d_in[13];
    const float* bl2     = (const float*)d_in[14];
    const float* wr2     = (const float*)d_in[15];
    const float* br2     = (const float*)d_in[16];
    const float* att2    = (const float*)d_in[17];
    const float* bias2   = (const float*)d_in[18];
    const float* q_w1    = (const float*)d_in[19];
    const float* q_b1    = (const float*)d_in[20];
    const float* q_w2    = (const float*)d_in[21];
    const float* q_b2    = (const float*)d_in[22];
    const float* v_w1    = (const float*)d_in[23];
    const float* v_b1    = (const float*)d_in[24];
    const float* v_w2    = (const float*)d_in[25];
    const float* v_b2    = (const float*)d_in[26];
    float* out = (float*)d_out;

    const int* esrc = eidx;
    const int* edst = eidx + EE;

    // ---- workspace layout (overlaid; ~115 MB total) ----
    float* p = (float*)d_ws;
    float* h1    = p; p += (size_t)NN * ENC_H;   // 10.24M f; later reused as xl2|xr2
    float* henc  = p; p += (size_t)NN * HID;     // 1.28M f
    float* xl1   = p; p += (size_t)NN * DD;      // 5.12M f; later reused as g2
    float* xr1   = p; p += (size_t)NN * DD;      // 5.12M f
    float* g1    = p; p += (size_t)NN * DD;      // 5.12M f
    float* score = p; p += (size_t)ETOT * HEADS; // 1.36M f
    float* feat  = p; p += (size_t)BB * CAT;
    float* qh    = p; p += (size_t)BB * DUEL_H;
    float* vh    = p; p += (size_t)BB * DUEL_H;
    int* ip = (int*)p;
    int* cnt     = ip; ip += NN;
    int* row_ptr = ip; ip += NN + 8;
    int* cursor  = ip; ip += NN;
    int* col     = ip; ip += ETOT;

    float* xl2 = h1;                      // overlay: encoder intermediate is dead
    float* xr2 = h1 + (size_t)NN * DD;
    float* g2  = xl1;                     // overlay: xl1 dead after layer-1 aggregate

    // ---- CSR build (shared by both GAT layers) ----
    zero_i32<<<(NN + 255) / 256, 256, 0, stream>>>(cnt, NN);
    count_edges<<<(ETOT + 255) / 256, 256, 0, stream>>>(edst, cnt);
    scan_counts<<<1, 1024, 0, stream>>>(cnt, row_ptr);
    copy_i32<<<(NN + 255) / 256, 256, 0, stream>>>(row_ptr, cursor, NN);
    fill_csr<<<(ETOT + 255) / 256, 256, 0, stream>>>(edst, cursor, col);

    // ---- encoder MLP ----
    {   // h1 = relu(x @ enc_w1 + b1)   M=20000 K=64 N=512
        int tiles = (NN / 16) * (ENC_H / 64);
        gemm_wmma_f32<<<(tiles + 3) / 4, 128, 0, stream>>>(x, enc_w1, enc_b1, h1, NN, ENC_H, IN_DIM, 1);
    }
    {   // henc = relu(h1 @ enc_w2 + b2)  M=20000 K=512 N=64
        int tiles = (NN / 16) * (HID / 64);
        gemm_wmma_f32<<<(tiles + 3) / 4, 128, 0, stream>>>(h1, enc_w2, enc_b2, henc, NN, HID, ENC_H, 1);
    }

    // ---- GAT layer 1 ----
    {
        int tiles = (NN / 16) * (DD / 64);
        gemm_wmma_f32<<<(tiles + 3) / 4, 128, 0, stream>>>(henc, wl1, bl1, xl1, NN, DD, HID, 0);
        gemm_wmma_f32<<<(tiles + 3) / 4, 128, 0, stream>>>(henc, wr1, br1, xr1, NN, DD, HID, 0);
    }
    edge_scores<<<(ETOT + 7) / 8, 256, 0, stream>>>(xl1, xr1, att1, esrc, edst, score);
    aggregate<<<(NN + 7) / 8, 256, 0, stream>>>(row_ptr, col, esrc, score, xl1, bias1, g1);

    // ---- GAT layer 2 ----
    {
        int tiles = (NN / 16) * (DD / 64);
        gemm_wmma_f32<<<(tiles + 3) / 4, 128, 0, stream>>>(g1, wl2, bl2, xl2, NN, DD, DD, 0);
        gemm_wmma_f32<<<(tiles + 3) / 4, 128, 0, stream>>>(g1, wr2, br2, xr2, NN, DD, DD, 0);
    }
    edge_scores<<<(ETOT + 7) / 8, 256, 0, stream>>>(xl2, xr2, att2, esrc, edst, score);
    aggregate<<<(NN + 7) / 8, 256, 0, stream>>>(row_ptr, col, esrc, score, xl2, bias2, g2);

    // ---- dueling head ----
    build_feat<<<(BB * CAT + 255) / 256, 256, 0, stream>>>(indices, henc, g1, g2, feat);
    {
        int tiles = (BB / 16) * (DUEL_H / 64);   // 16 tiles
        gemm_wmma_f32<<<(tiles + 3) / 4, 128, 0, stream>>>(feat, q_w1, q_b1, qh, BB, DUEL_H, CAT, 1);
        gemm_wmma_f32<<<(tiles + 3) / 4, 128, 0, stream>>>(feat, v_w1, v_b1, vh, BB, DUEL_H, CAT, 1);
    }
    head_final<<<1, 128, 0, stream>>>(qh, vh, q_w2, q_b2, v_w2, v_b2, out);
}